// PaddedLobPredModel_51453708206537
// MI455X (gfx1250) — compile-verified
//
#include <hip/hip_runtime.h>

// EWMA scan: y_t = decay*y_{t-1} + alpha*x_t over (L=32768, D=1024), f32.
// Bandwidth-bound (0.25 FLOP/byte). Strategy: chunked parallel scan.
//   Pass 1: per-(chunk,col) zero-carry local scan -> chunk end values
//   Pass 2: per-col serial combine of 64 chunk ends -> per-chunk carry-ins + last row
//   Pass 3: per-wave 16-col tile; 16 timesteps at once via T(16x16) @ X(16x16)
//           using 4 chained V_WMMA_F32_16X16X4_F32, carry fixed up with decay^(m+1),
//           new carry broadcast from D row 15 via wave32 shuffle.

#define L_T   32768
#define D_C   1024
#define CHUNK 512
#define NCHUNK (L_T / CHUNK)   // 64 chunks

typedef __attribute__((ext_vector_type(2))) float v2f;
typedef __attribute__((ext_vector_type(8))) float v8f;

// ---------------- Pass 1: local (zero-carry) chunk end values ----------------
// gid -> (chunk, group of 4 columns). float4 loads: fully coalesced, 4-way ILP.
__global__ void __launch_bounds__(256)
ewma_chunk_ends(const float* __restrict__ data,
                const float* __restrict__ alpha_p,
                float* __restrict__ end_local) {
    const float alpha = *alpha_p;
    const float decay = 1.0f - alpha;
    const int gid = blockIdx.x * blockDim.x + threadIdx.x;    // 0 .. 64*256-1
    const int c   = gid / (D_C / 4);
    const int c4  = gid % (D_C / 4);

    const float4* row = (const float4*)data + (size_t)c * CHUNK * (D_C / 4) + c4;
    float4 y = make_float4(0.f, 0.f, 0.f, 0.f);
#pragma unroll 8
    for (int r = 0; r < CHUNK; ++r) {
        float4 x = row[(size_t)r * (D_C / 4)];
        y.x = fmaf(decay, y.x, alpha * x.x);
        y.y = fmaf(decay, y.y, alpha * x.y);
        y.z = fmaf(decay, y.z, alpha * x.z);
        y.w = fmaf(decay, y.w, alpha * x.w);
    }
    ((float4*)end_local)[(size_t)c * (D_C / 4) + c4] = y;
}

// ---------------- Pass 2: combine chunk ends -> carry-ins + last row ---------
__global__ void __launch_bounds__(256)
ewma_chunk_carries(const float* __restrict__ first_offset,
                   const float* __restrict__ alpha_p,
                   const float* __restrict__ end_local,
                   float* __restrict__ carry_in,
                   float* __restrict__ d_last) {
    const float alpha = *alpha_p;
    const float decay = 1.0f - alpha;
    // decay^CHUNK, CHUNK = 512 = 2^9 -> 9 squarings
    float dpc = decay;
#pragma unroll
    for (int i = 0; i < 9; ++i) dpc = dpc * dpc;

    const int col = blockIdx.x * blockDim.x + threadIdx.x;    // 0..1023
    float carry = first_offset[col];
    for (int c = 0; c < NCHUNK; ++c) {
        carry_in[c * D_C + col] = carry;
        carry = fmaf(dpc, carry, end_local[c * D_C + col]);
    }
    d_last[col] = carry;   // == y_{L-1}
}

// ---------------- Pass 3: full output via WMMA -------------------------------
// Wave w -> (chunk = w/64, column tile = w%64). Per 16-row block:
//   D(16x16) = T(16x16) @ X(16x16)   via 4x V_WMMA_F32_16X16X4_F32 (K=4 each)
//   out[m][n] = D[m][n] + decay^(m+1) * carry[n]
//   carry'[n] = out[15][n]  (D-layout: VGPR7, lanes 16..31) -> __shfl broadcast
__global__ void __launch_bounds__(256)
ewma_wmma_out(const float* __restrict__ data,
              const float* __restrict__ alpha_p,
              const float* __restrict__ carry_in,
              float* __restrict__ out) {
    const float alpha = *alpha_p;
    const float decay = 1.0f - alpha;

    const int lane  = threadIdx.x & 31;
    const int w     = blockIdx.x * (blockDim.x >> 5) + (threadIdx.x >> 5);
    const int chunk = w >> 6;            // / (D_C/16)
    const int ctile = w & 63;            // % (D_C/16)
    const int m     = lane & 15;         // matrix row for A, column-in-tile for B/C/D
    const int hi    = lane >> 4;         // 0: lanes 0-15, 1: lanes 16-31
    const int koff  = hi * 2;            // K offset inside a 4-slice (ISA A/B layout)
    const int n     = ctile * 16 + m;    // global column

    // decay^i, i = 0..15 (max exponent needed in T and dp)
    float pw[16];
    pw[0] = 1.0f;
#pragma unroll
    for (int i = 1; i < 16; ++i) pw[i] = pw[i - 1] * decay;

    // A slices: A[kk] holds T[m][4kk+koff], T[m][4kk+koff+1]
    // T[m][k] = alpha * decay^(m-k) for k<=m else 0 (lower-triangular scan matrix)
    v2f A[4];
#pragma unroll
    for (int kk = 0; kk < 4; ++kk) {
        const int k0 = 4 * kk + koff;
        const int k1 = k0 + 1;
        A[kk].x = (k0 <= m) ? alpha * pw[m - k0] : 0.0f;
        A[kk].y = (k1 <= m) ? alpha * pw[m - k1] : 0.0f;
    }

    // carry factors per D-VGPR v: decay^(m_d+1), m_d = v + hi*8
    float dp[8];
    dp[0] = decay * (hi ? pw[8] : 1.0f);
#pragma unroll
    for (int v = 1; v < 8; ++v) dp[v] = dp[v - 1] * decay;

    float carry = carry_in[chunk * D_C + n];
    const int row_base = chunk * CHUNK;

    for (int blk = 0; blk < CHUNK / 16; ++blk) {
        const int r0 = row_base + blk * 16;

        // B slices (K x N = 4x16): VGPR j, lanes 0-15 -> K=j, lanes 16-31 -> K=j+2
        v2f B[4];
#pragma unroll
        for (int kk = 0; kk < 4; ++kk) {
            const int k0 = 4 * kk + koff;
            B[kk].x = data[(size_t)(r0 + k0) * D_C + n];
            B[kk].y = data[(size_t)(r0 + k0 + 1) * D_C + n];
        }

        v8f c = {};
#pragma unroll
        for (int kk = 0; kk < 4; ++kk) {
            c = __builtin_amdgcn_wmma_f32_16x16x4_f32(
                    /*neg_a=*/false, A[kk], /*neg_b=*/false, B[kk],
                    /*c_mod=*/(short)0, c, /*reuse_a=*/false, /*reuse_b=*/false);
        }

        // carry fix-up + store (D layout: VGPR v -> row v + hi*8, col = lane%16)
        float o7;
#pragma unroll
        for (int v = 0; v < 8; ++v) {
            float o = fmaf(dp[v], carry, c[v]);
            out[(size_t)(r0 + v + hi * 8) * D_C + n] = o;
            if (v == 7) o7 = o;
        }
        // new carry = out row 15 (held by lane 16+col at VGPR 7)
        carry = __shfl(o7, 16 + m, 32);
    }
}

// ---------------- Launcher ---------------------------------------------------
extern "C" void kernel_launch(void* const* d_in, const int* in_sizes, int n_in,
                              void* d_out, int out_size, void* d_ws, size_t ws_size,
                              hipStream_t stream) {
    (void)in_sizes; (void)n_in; (void)out_size; (void)ws_size;

    const float* data    = (const float*)d_in[0];   // (L, D) f32
    const float* first   = (const float*)d_in[1];   // (1, D) f32
    const float* alpha_p = (const float*)d_in[2];   // scalar f32

    float* out    = (float*)d_out;                  // (L, D) then (1, D)
    float* d_last = out + (size_t)L_T * D_C;

    float* end_local = (float*)d_ws;                 // NCHUNK*D floats (256 KB)
    float* carries   = end_local + NCHUNK * D_C;     // NCHUNK*D floats (256 KB)

    // Pass 1: 64 chunks * 256 float4-columns = 16384 threads
    ewma_chunk_ends<<<(NCHUNK * (D_C / 4)) / 256, 256, 0, stream>>>(
        data, alpha_p, end_local);

    // Pass 2: 1024 columns
    ewma_chunk_carries<<<D_C / 256, 256, 0, stream>>>(
        first, alpha_p, end_local, carries, d_last);

    // Pass 3: 64 chunks * 64 col-tiles = 4096 waves = 131072 threads
    ewma_wmma_out<<<(NCHUNK * (D_C / 16) * 32) / 256, 256, 0, stream>>>(
        data, alpha_p, carries, out);
}